// GridPool_463856468481
// MI455X (gfx1250) — compile-verified
//
#include <hip/hip_runtime.h>
#include <hip/hip_bf16.h>

#define BATCH 8
#define NPTS 65536
#define CIN 64
#define COUT 128
#define VOX 32768
#define GRIDSZ 0.08f
#define LN_EPS 1e-5f

typedef __attribute__((ext_vector_type(2))) float v2f;
typedef __attribute__((ext_vector_type(8))) float v8f;

// LDS byte offset of a generic pointer to __shared__: the LDS aperture
// occupies addr[63:32] only (ISA 10.2 aperture check), so the low 32 bits
// are the segment-relative LDS offset.
__device__ __forceinline__ unsigned lds_off_of(const void* p) {
    return (unsigned)(unsigned long long)p;
}

// CDNA5 async DMA: GLOBAL_LOAD_ASYNC_TO_LDS_B128 (ASYNCcnt-tracked).
// Each active lane copies 16B from its global address to its LDS address.
__device__ __forceinline__ void async_g2l_b128(unsigned lds_byte_off, const void* gptr) {
    asm volatile("global_load_async_to_lds_b128 %0, %1, off"
                 :: "v"(lds_byte_off), "v"((unsigned long long)gptr)
                 : "memory");
}
__device__ __forceinline__ void wait_async0() {
    asm volatile("s_wait_asynccnt 0x0" ::: "memory");
}

// ---------------------------------------------------------------- utilities
__global__ void k_init_minmax(int* minmax) {
    int i = blockIdx.x * blockDim.x + threadIdx.x;   // BATCH*6
    if (i >= BATCH * 6) return;
    int axis = i % 6;
    minmax[i] = (axis < 3) ? 0x7fffffff : (int)0x80000000;
}

// ------------------------------------------------- per-batch coord min/max
__global__ void k_minmax(const float* __restrict__ xyz, int* __restrict__ minmax) {
    __shared__ int smin[3], smax[3];
    if (threadIdx.x < 3) { smin[threadIdx.x] = 0x7fffffff; smax[threadIdx.x] = (int)0x80000000; }
    __syncthreads();
    int i = blockIdx.x * blockDim.x + threadIdx.x;   // 256|NPTS -> b uniform per block
    int b = i / NPTS;
    const float* p = xyz + (long)i * 3;
    #pragma unroll
    for (int a = 0; a < 3; ++a) {
        int c = (int)floorf(p[a] / GRIDSZ);
        atomicMin(&smin[a], c);
        atomicMax(&smax[a], c);
    }
    __syncthreads();
    if (threadIdx.x < 3) {
        atomicMin(&minmax[b * 6 + threadIdx.x], smin[threadIdx.x]);
        atomicMax(&minmax[b * 6 + 3 + threadIdx.x], smax[threadIdx.x]);
    }
}

// ---------------------------------------------- linearized keys + occupancy
__global__ void k_keys(const float* __restrict__ xyz, const int* __restrict__ minmax,
                       int* __restrict__ keys, int* __restrict__ voxmap) {
    int i = blockIdx.x * blockDim.x + threadIdx.x;
    if (i >= BATCH * NPTS) return;
    int b = i / NPTS;
    int m0 = minmax[b * 6 + 0], m1 = minmax[b * 6 + 1], m2 = minmax[b * 6 + 2];
    int d1 = minmax[b * 6 + 4] - m1 + 1;
    int d2 = minmax[b * 6 + 5] - m2 + 1;
    const float* p = xyz + (long)i * 3;
    int c0 = (int)floorf(p[0] / GRIDSZ) - m0;
    int c1 = (int)floorf(p[1] / GRIDSZ) - m1;
    int c2 = (int)floorf(p[2] / GRIDSZ) - m2;
    int key = c0 * d1 * d2 + c1 * d2 + c2;
    key = min(max(key, 0), VOX - 1);
    keys[i] = key;
    voxmap[b * VOX + key] = 1;   // racing stores of identical value: benign
}

// ------------------- exclusive scan of occupancy -> ascending-key cluster ids
__global__ __launch_bounds__(1024) void k_scan(const int* __restrict__ voxmap,
                                               int* __restrict__ cmap, int* __restrict__ Ms) {
    __shared__ int sd[1024];
    __shared__ int carry;
    int b = blockIdx.x, t = threadIdx.x;
    if (t == 0) carry = 0;
    __syncthreads();
    for (int base = 0; base < VOX; base += 1024) {
        int occ = voxmap[b * VOX + base + t];
        sd[t] = occ;
        __syncthreads();
        #pragma unroll
        for (int off = 1; off < 1024; off <<= 1) {
            int v = (t >= off) ? sd[t - off] : 0;
            __syncthreads();
            sd[t] += v;
            __syncthreads();
        }
        int incl = sd[t];
        cmap[b * VOX + base + t] = occ ? (carry + incl - occ) : -1;
        int blockTotal = sd[1023];
        __syncthreads();
        if (t == 0) carry += blockTotal;
        __syncthreads();
    }
    if (t == 0) Ms[b] = carry;
}

__global__ void k_maxM(const int* __restrict__ Ms, float* __restrict__ out_maxM) {
    int m = 0;
    for (int b = 0; b < BATCH; ++b) m = max(m, Ms[b]);
    *out_maxM = (float)m;
}

// --------------------------- per-point: cluster id out + xyz/count accumulate
__global__ void k_accum_point(const float* __restrict__ xyz, const int* __restrict__ keys,
                              const int* __restrict__ cmap, int* __restrict__ cids,
                              float* __restrict__ clusterOut, float* __restrict__ xyzSum,
                              float* __restrict__ counts) {
    int i = blockIdx.x * blockDim.x + threadIdx.x;
    if (i >= BATCH * NPTS) return;
    int b = i / NPTS;
    int cid = cmap[b * VOX + keys[i]];
    if (cid < 0) cid = 0;
    cids[i] = cid;
    clusterOut[i] = (float)cid;
    long r = (long)b * NPTS + cid;
    atomicAdd(&counts[r], 1.0f);
    const float* p = xyz + (long)i * 3;
    atomicAdd(&xyzSum[r * 3 + 0], p[0]);
    atomicAdd(&xyzSum[r * 3 + 1], p[1]);
    atomicAdd(&xyzSum[r * 3 + 2], p[2]);
}

__global__ void k_accum_feat(const float* __restrict__ feats, const int* __restrict__ cids,
                             float* __restrict__ featSum) {
    long i = (long)blockIdx.x * blockDim.x + threadIdx.x;  // BATCH*NPTS*CIN
    if (i >= (long)BATCH * NPTS * CIN) return;
    long p = i >> 6;            // point index
    int  c = (int)(i & 63);
    long b = p / NPTS;
    int cid = cids[p];
    atomicAdd(&featSum[((long)b * NPTS + cid) * CIN + c], feats[i]);
}

__global__ void k_div_feat(float* __restrict__ featSum, const float* __restrict__ counts) {
    long i = (long)blockIdx.x * blockDim.x + threadIdx.x;  // BATCH*NPTS*CIN
    if (i >= (long)BATCH * NPTS * CIN) return;
    float inv = 1.0f / fmaxf(counts[i >> 6], 1.0f);
    featSum[i] *= inv;
}

__global__ void k_div_xyz(float* __restrict__ xyzSum, const float* __restrict__ counts) {
    long i = (long)blockIdx.x * blockDim.x + threadIdx.x;  // BATCH*NPTS*3
    if (i >= (long)BATCH * NPTS * 3) return;
    float inv = 1.0f / fmaxf(counts[i / 3], 1.0f);
    xyzSum[i] *= inv;
}

// --------------------------------------- fused (BN,64)x(64,128)+b + LayerNorm
// One wave = 16x128 output tile via 8 accumulators of V_WMMA_F32_16X16X4_F32.
// A (per-wave 16x64 tile) staged via GLOBAL_LOAD_ASYNC_TO_LDS_B128 (ASYNCcnt
// DMA). W staged once per block in a k-pair-interleaved layout:
//     sW[(k/2)*LDWP + 2n + (k&1)]
// so each WMMA b-fragment {W[K][n], W[K+1][n]} (K even) is ONE contiguous
// 8B-aligned LDS element -> single ds_load_b64 into an even VGPR pair, no
// re-pairing v_movs. LDWP=288 (mod 64 == 32): kh=0 lanes hit banks 0..31,
// kh=2 lanes hit banks 32..63 -> conflict-free.
// LDA=68: row stride 272B (16B-aligned); 68%64=4 -> lane-half bank split.
__global__ __launch_bounds__(256) void k_gemm_ln(const float* __restrict__ A,
                                                 const float* __restrict__ W,
                                                 const float* __restrict__ bias,
                                                 const float* __restrict__ gamma,
                                                 const float* __restrict__ beta,
                                                 float* __restrict__ out) {
    const int LDA  = CIN + 4;                        // 68 floats
    const int LDWP = 2 * COUT + 32;                  // 288 floats per k-pair row
    __shared__ __align__(16) float sA[8 * 16 * (CIN + 4)];       // 34816 B
    __shared__ __align__(16) float sW[(CIN / 2) * (2 * COUT + 32)]; // 36864 B
    const int wave = threadIdx.x >> 5;
    const int lane = threadIdx.x & 31;
    const int lane15 = lane & 15;
    const int kh = (lane >> 4) * 2;                  // K sub-offset per lane half
    const long rowBase = (long)blockIdx.x * 128 + wave * 16;

    // ---- async-DMA stage of this wave's 16x64 A tile: 256 x b128
    float* myA = sA + wave * 16 * LDA;
    const float* gA = A + rowBase * CIN;
    #pragma unroll
    for (int i = 0; i < 8; ++i) {
        int q = i * 32 + lane;                       // float4 index 0..255
        int r = q >> 4, c4 = q & 15;
        async_g2l_b128(lds_off_of(&myA[r * LDA + c4 * 4]), gA + q * 4);
    }

    // ---- stage W k-pair-interleaved: 32 row-pairs x 32 float4-cols
    #pragma unroll
    for (int i = 0; i < 4; ++i) {
        int item = i * 256 + threadIdx.x;            // 0..1023
        int rp = item >> 5;                          // k-pair 0..31
        int n4 = (item & 31) * 4;                    // col base
        const float4 w0 = *(const float4*)(W + (2 * rp + 0) * COUT + n4);
        const float4 w1 = *(const float4*)(W + (2 * rp + 1) * COUT + n4);
        v2f* dst = (v2f*)&sW[rp * LDWP + 2 * n4];
        dst[0] = (v2f){w0.x, w1.x};
        dst[1] = (v2f){w0.y, w1.y};
        dst[2] = (v2f){w0.z, w1.z};
        dst[3] = (v2f){w0.w, w1.w};
    }

    wait_async0();
    __syncthreads();

    v8f acc[8];
    #pragma unroll
    for (int t = 0; t < 8; ++t) acc[t] = (v8f){0.f, 0.f, 0.f, 0.f, 0.f, 0.f, 0.f, 0.f};

    // ---- K = 64 as 16 steps of k=4; fragments are single b64 LDS loads
    const float* a_ptr = myA + lane15 * LDA + kh;                    // A[M][K..K+1]
    const float* w_ptr = &sW[(lane >> 4) * LDWP + lane15 * 2];       // {W[K][n],W[K+1][n]}
    #pragma unroll
    for (int kk = 0; kk < 16; ++kk) {
        v2f a = *(const v2f*)(a_ptr + kk * 4);
        #pragma unroll
        for (int t = 0; t < 8; ++t) {
            v2f bb = *(const v2f*)(w_ptr + kk * 2 * LDWP + t * 32);
            acc[t] = __builtin_amdgcn_wmma_f32_16x16x4_f32(
                false, a, false, bb, (short)0, acc[t], false, false);
        }
    }

    // ---- bias + LayerNorm fused in registers (C layout: vgpr v -> rows v/v+8)
    float bi[8], g[8], be[8];
    #pragma unroll
    for (int t = 0; t < 8; ++t) {
        int col = t * 16 + lane15;
        bi[t] = bias[col]; g[t] = gamma[col]; be[t] = beta[col];
    }
    float s[8], sq[8];
    #pragma unroll
    for (int v = 0; v < 8; ++v) {
        float ss = 0.f, qq = 0.f;
        #pragma unroll
        for (int t = 0; t < 8; ++t) {
            float h = acc[t][v] + bi[t];
            acc[t][v] = h;
            ss += h; qq += h * h;
        }
        s[v] = ss; sq[v] = qq;
    }
    #pragma unroll
    for (int off = 1; off < 16; off <<= 1) {         // butterfly within each 16-lane half
        #pragma unroll
        for (int v = 0; v < 8; ++v) {
            s[v]  += __shfl_xor(s[v],  off, 32);
            sq[v] += __shfl_xor(sq[v], off, 32);
        }
    }
    const int rowOff = (lane >= 16) ? 8 : 0;
    #pragma unroll
    for (int v = 0; v < 8; ++v) {
        float mu  = s[v] * (1.0f / COUT);
        float var = sq[v] * (1.0f / COUT) - mu * mu;
        float rs  = rsqrtf(var + LN_EPS);
        float* op = out + (rowBase + v + rowOff) * COUT;
        #pragma unroll
        for (int t = 0; t < 8; ++t)
            op[t * 16 + lane15] = (acc[t][v] - mu) * rs * g[t] + be[t];
    }
}

// ---------------------------------------------------------------------------
extern "C" void kernel_launch(void* const* d_in, const int* in_sizes, int n_in,
                              void* d_out, int out_size, void* d_ws, size_t ws_size,
                              hipStream_t stream) {
    const float* xyz   = (const float*)d_in[0];
    const float* feats = (const float*)d_in[1];
    const float* W     = (const float*)d_in[2];
    const float* bias  = (const float*)d_in[3];
    const float* gamma = (const float*)d_in[4];
    const float* beta  = (const float*)d_in[5];

    float* out_xyz     = (float*)d_out;                          // (B,N,3)
    float* out_feats   = out_xyz   + (size_t)BATCH * NPTS * 3;   // (B,N,128)
    float* out_cluster = out_feats + (size_t)BATCH * NPTS * COUT;// (B,N)
    float* out_maxM    = out_cluster + (size_t)BATCH * NPTS;     // scalar

    char* w = (char*)d_ws;
    float* featSum = (float*)w;  w += (size_t)BATCH * NPTS * CIN * 4;
    float* counts  = (float*)w;  w += (size_t)BATCH * NPTS * 4;
    int*   keys    = (int*)w;    w += (size_t)BATCH * NPTS * 4;
    int*   cids    = (int*)w;    w += (size_t)BATCH * NPTS * 4;
    int*   voxmap  = (int*)w;    w += (size_t)BATCH * VOX * 4;
    int*   cmap    = (int*)w;    w += (size_t)BATCH * VOX * 4;
    int*   minmax  = (int*)w;    w += BATCH * 6 * 4;
    int*   Ms      = (int*)w;    w += BATCH * 4;

    // zero accumulators (graph-capture-safe async memsets)
    hipMemsetAsync(featSum, 0, (size_t)BATCH * NPTS * CIN * 4, stream);
    hipMemsetAsync(counts,  0, (size_t)BATCH * NPTS * 4, stream);
    hipMemsetAsync(voxmap,  0, (size_t)BATCH * VOX * 4, stream);
    hipMemsetAsync(out_xyz, 0, (size_t)BATCH * NPTS * 3 * 4, stream);

    const int BN = BATCH * NPTS;
    k_init_minmax<<<1, 64, 0, stream>>>(minmax);
    k_minmax<<<BN / 256, 256, 0, stream>>>(xyz, minmax);
    k_keys<<<BN / 256, 256, 0, stream>>>(xyz, minmax, keys, voxmap);
    k_scan<<<BATCH, 1024, 0, stream>>>(voxmap, cmap, Ms);
    k_maxM<<<1, 1, 0, stream>>>(Ms, out_maxM);
    k_accum_point<<<BN / 256, 256, 0, stream>>>(xyz, keys, cmap, cids, out_cluster,
                                                out_xyz, counts);
    k_accum_feat<<<(int)(((long)BN * CIN) / 256), 256, 0, stream>>>(feats, cids, featSum);
    k_div_feat<<<(int)(((long)BN * CIN) / 256), 256, 0, stream>>>(featSum, counts);
    k_div_xyz<<<(BN * 3 + 255) / 256, 256, 0, stream>>>(out_xyz, counts);
    k_gemm_ln<<<BN / 128, 256, 0, stream>>>(featSum, W, bias, gamma, beta, out_feats);
}